// GCNLayer_86483461472648
// MI455X (gfx1250) — compile-verified
//
#include <hip/hip_runtime.h>
#include <hip/hip_bf16.h>

typedef float v2f __attribute__((ext_vector_type(2)));
typedef float v8f __attribute__((ext_vector_type(8)));

#define IN_DIM  128
#define OUT_DIM 128

// ---------------------------------------------------------------- kernel 1
__global__ void gcn_zero_deg(float* __restrict__ deg, int nn) {
    int i = blockIdx.x * blockDim.x + threadIdx.x;
    if (i < nn) deg[i] = 0.0f;
}

// ---------------------------------------------------------------- kernel 2
__global__ void gcn_degree(const int* __restrict__ hs, float* __restrict__ deg, int ne) {
    int e = blockIdx.x * blockDim.x + threadIdx.x;
    if (e < ne) atomicAdd(&deg[hs[e]], 1.0f);
}

// ---------------------------------------------------------------- kernel 3
// scaled[i] = deg[row]^-1/2 * feats[i]; also zero agg[i]
__global__ void gcn_scale_zero(const float* __restrict__ feats,
                               const float* __restrict__ deg,
                               float* __restrict__ scaled,
                               float* __restrict__ agg, int total) {
    int i = blockIdx.x * blockDim.x + threadIdx.x;
    if (i >= total) return;
    int row = i >> 7;                    // / IN_DIM
    float s = rsqrtf(deg[row]);
    scaled[i] = s * feats[i];
    agg[i]    = 0.0f;
}

// ---------------------------------------------------------------- kernel 4
// SpMM scatter: one wave per edge, one float4 per lane (32*4 = 128 dims).
// scaled and agg are both ~51 MB -> resident in the 192 MB L2, so the random
// gather/scatter + f32 atomics stay on-die.
__global__ void gcn_spmm(const int* __restrict__ hs, const int* __restrict__ ts,
                         const float* __restrict__ scaled, float* __restrict__ agg,
                         int ne) {
    long long idx = (long long)blockIdx.x * blockDim.x + threadIdx.x;
    int e = (int)(idx >> 5);
    if (e >= ne) return;
    int lane = (int)(idx & 31);
    int h = hs[e];
    int t = ts[e];
    const float4 v = *((const float4*)(scaled + (size_t)t * IN_DIM) + lane);
    float* dst = agg + (size_t)h * IN_DIM + lane * 4;
    atomicAdd(dst + 0, v.x);
    atomicAdd(dst + 1, v.y);
    atomicAdd(dst + 2, v.z);
    atomicAdd(dst + 3, v.w);
}

// ---------------------------------------------------------------- kernel 5
// out = relu( (deg^-1/2 * agg) @ W^T ) with fp32 WMMA 16x16x4.
// One wave computes a 16-row strip x all 128 output cols.
// A frag layout (ISA 7.12.2, 32-bit A 16x4): lanes 0-15 carry M=lane,
//   VGPR0/1 = K0,K1; lanes 16-31 carry same M with K2,K3
//   -> lane loads float2 at msg[row] + 4*j + 2*(lane>>4).
// B frag (4x16): VGPR0 = rows K0 (lanes 0-15) / K2 (lanes 16-31), VGPR1 = K1/K3
//   -> lane loads float2 at W[n0 + lane%16] + 4*j + 2*(lane>>4).
// C/D (16x16, 8 VGPRs): VGPR r = row r (lanes 0-15) / row r+8 (lanes 16-31).
__global__ __launch_bounds__(256) void gcn_gemm_relu(
        const float* __restrict__ agg, const float* __restrict__ deg,
        const float* __restrict__ W, float* __restrict__ out, int nn) {
    int wave = (blockIdx.x * blockDim.x + threadIdx.x) >> 5;
    int lane = threadIdx.x & 31;
    int ntiles = (nn + 15) >> 4;
    if (wave >= ntiles) return;               // wave-uniform: EXEC stays all-1s

    int m  = lane & 15;
    int kh = lane >> 4;
    int row = wave * 16 + m;
    if (row >= nn) row = nn - 1;              // clamp (N_N % 16 == 0 normally)
    float s = rsqrtf(deg[row]);

    // Hoist the full A strip: 32 k-steps x float2 = 64 VGPRs, scaled by s.
    v2f a[32];
    const float* ap = agg + (size_t)row * IN_DIM + 2 * kh;
#pragma unroll
    for (int j = 0; j < 32; ++j) {
        v2f t = *(const v2f*)(ap + 4 * j);
        a[j] = t * s;
    }

#pragma unroll
    for (int nt = 0; nt < 8; ++nt) {          // 8 column tiles of 16
        v8f c = {};
        const float* bp = W + (size_t)(nt * 16 + m) * IN_DIM + 2 * kh;
#pragma unroll
        for (int j = 0; j < 32; ++j) {        // K = 128 in steps of 4
            v2f b = *(const v2f*)(bp + 4 * j);
            c = __builtin_amdgcn_wmma_f32_16x16x4_f32(
                    /*neg_a=*/false, a[j], /*neg_b=*/false, b,
                    /*c_mod=*/(short)0, c, /*reuse_a=*/false, /*reuse_b=*/false);
        }
        int baseM = wave * 16 + kh * 8;
#pragma unroll
        for (int r = 0; r < 8; ++r) {
            int om = baseM + r;
            if (om < nn)
                out[(size_t)om * OUT_DIM + nt * 16 + m] = fmaxf(c[r], 0.0f);
        }
    }
}

// ---------------------------------------------------------------- launch
extern "C" void kernel_launch(void* const* d_in, const int* in_sizes, int n_in,
                              void* d_out, int out_size, void* d_ws, size_t ws_size,
                              hipStream_t stream) {
    const float* feats  = (const float*)d_in[0];
    const int*   edges  = (const int*)d_in[1];   // [2, N_E]: hs then ts
    const float* weight = (const float*)d_in[2]; // [OUT_DIM, IN_DIM]

    const int nn = in_sizes[0] / IN_DIM;
    const int ne = in_sizes[1] / 2;
    const int* hs = edges;
    const int* ts = edges + ne;

    // workspace layout (floats): deg[nn] | scaled[nn*128] | agg[nn*128]
    float* deg    = (float*)d_ws;
    float* scaled = deg + nn;
    float* agg    = scaled + (size_t)nn * IN_DIM;

    float* out = (float*)d_out;

    const int B = 256;

    gcn_zero_deg<<<(nn + B - 1) / B, B, 0, stream>>>(deg, nn);

    gcn_degree<<<(ne + B - 1) / B, B, 0, stream>>>(hs, deg, ne);

    const int total = nn * IN_DIM;
    gcn_scale_zero<<<(total + B - 1) / B, B, 0, stream>>>(feats, deg, scaled, agg, total);

    const long long spmm_threads = (long long)ne * 32;
    gcn_spmm<<<(int)((spmm_threads + B - 1) / B), B, 0, stream>>>(hs, ts, scaled, agg, ne);

    const int ntiles = (nn + 15) / 16;
    const int gemm_blocks = (ntiles * 32 + B - 1) / B;  // 8 waves per block
    gcn_gemm_relu<<<gemm_blocks, B, 0, stream>>>(agg, deg, weight, out, nn);
}